// ISDALoss_23072564314191
// MI455X (gfx1250) — compile-verified
//
#include <hip/hip_runtime.h>
#include <math.h>

// ---------------------------------------------------------------------------
// ISDA loss pipeline for gfx1250 (MI455X).
// Problem sizes fixed by setup_inputs(): N=512, A=2048, C=1000, f32.
// ---------------------------------------------------------------------------

typedef __attribute__((ext_vector_type(2))) float v2f;
typedef __attribute__((ext_vector_type(8))) float v8f;
typedef __attribute__((ext_vector_type(4))) int   v4i;   // async b128 payload

#define N_S   512
#define A_F   2048
#define C_CLS 1000

// GEMM tiling: block = 256 threads = 8 wave32s.
// Block tile 64(M) x 128(N), K-chunk 32. Wave tile 16(M) x 64(N).
#define BM 64
#define BN 128
#define BK 32
// LDS row pitch 36 floats = 144B: 16B-aligned for async b128 copies and
// bank-conflict-free for the v2f fragment reads (same-half collision needs
// row-delta 16; cross-half needs 18*d == 1 mod 32 -> impossible).
#define LDT 36

// ---- CDNA5 async global->LDS path (guarded; falls back to sync staging) ----
#if __has_builtin(__builtin_amdgcn_global_load_async_to_lds_b128)
#define USE_ASYNC_LDS 1
#else
#define USE_ASYNC_LDS 0
#endif

__device__ __forceinline__ void wait_async_zero() {
#if __has_builtin(__builtin_amdgcn_s_wait_asynccnt)
    __builtin_amdgcn_s_wait_asynccnt(0);
#else
    asm volatile("s_wait_asynccnt 0x0" ::: "memory");
#endif
}

// ------------------------------ utility ------------------------------------

__global__ void fillzero_kernel(float* __restrict__ p, int n) {
    int i = blockIdx.x * blockDim.x + threadIdx.x;
    int stride = gridDim.x * blockDim.x;
    for (; i < n; i += stride) p[i] = 0.0f;
}

// ------------------------- update_cv statistics ----------------------------

__global__ void counts_kernel(const int* __restrict__ labels,
                              float* __restrict__ counts) {
    int n = blockIdx.x * blockDim.x + threadIdx.x;
    if (n < N_S) atomicAdd(&counts[labels[n]], 1.0f);
}

__global__ void sums_kernel(const float* __restrict__ feat,
                            const int* __restrict__ labels,
                            float* __restrict__ sums) {
    int idx = blockIdx.x * blockDim.x + threadIdx.x;   // N*A = 1,048,576
    if (idx >= N_S * A_F) return;
    int n = idx / A_F;
    int a = idx - n * A_F;
    atomicAdd(&sums[(long long)labels[n] * A_F + a], feat[idx]);
}

__global__ void sq_kernel(const float* __restrict__ feat,
                          const int* __restrict__ labels,
                          const float* __restrict__ counts,
                          const float* __restrict__ sums,
                          float* __restrict__ sq) {
    int idx = blockIdx.x * blockDim.x + threadIdx.x;
    int stride = gridDim.x * blockDim.x;
    for (; idx < N_S * A_F; idx += stride) {
        int n = idx / A_F;
        int a = idx - n * A_F;
        int c = labels[n];
        float cnt = counts[c];
        float den = (cnt == 0.0f) ? 1.0f : cnt;
        float m = sums[(long long)c * A_F + a] / den;
        float d = feat[idx] - m;
        atomicAdd(&sq[(long long)c * A_F + a], d * d);
    }
}

// ave_new currently holds per-class sums, cov_new holds squared-diff sums;
// finalize them in place into the output regions.
__global__ void finalize_cv_kernel(const float* __restrict__ counts,
                                   const float* __restrict__ ave_old,
                                   const float* __restrict__ cov_old,
                                   const float* __restrict__ amt_old,
                                   float* __restrict__ ave_new,
                                   float* __restrict__ cov_new,
                                   float* __restrict__ amt_new) {
    int idx = blockIdx.x * blockDim.x + threadIdx.x;
    int stride = gridDim.x * blockDim.x;
    for (; idx < C_CLS * A_F; idx += stride) {
        int c = idx / A_F;
        int a = idx - c * A_F;
        float cnt = counts[c];
        float den = (cnt == 0.0f) ? 1.0f : cnt;
        float aveCA = ave_new[idx] / den;   // sums
        float var   = cov_new[idx] / den;   // sq sums
        float amt = amt_old[c];
        float s = cnt + amt;
        float w = (s == 0.0f) ? 0.0f : (cnt / s);   // == nan_to_num(cnt/s)
        float d0 = ave_old[idx] - aveCA;
        float addl = w * (1.0f - w) * d0 * d0;
        cov_new[idx] = cov_old[idx] * (1.0f - w) + var * w + addl;
        ave_new[idx] = ave_old[idx] * (1.0f - w) + aveCA * w;
        if (a == 0) amt_new[c] = amt + cnt;
    }
}

// ------------------------------ WMMA GEMM ----------------------------------
// OUT[m, n] = sum_k Amat[m, k] * Bmat[n, k]   (A @ B^T, both row-major)
// mode 0: out = acc + bias[n]
// mode 1: out = ybase + 0.5*ratio*(acc + t3[m])
//
// f32 WMMA 16x16x4 fragment layout (ISA 7.12.2):
//   A/B:  lane l holds element [row = l&15, k = k0 + 2*(l>>4) + j] in vgpr j
//   C/D:  vgpr r at lane l holds D[M = r + 8*(l>>4), N = l&15]
//
// OOB B rows (output cols >= Cdim) are CLAMPED, not zeroed: a WMMA output
// column depends only on its own B fragment row, and the epilogue never
// stores cols >= Cdim, so the resulting garbage columns are dead values.

#if USE_ASYNC_LDS
#define NBUF 2
// Copy a (rows x BK) f32 tile into LDS via async b128 (16B per lane-op).
__device__ __forceinline__ void stage_async(const float* __restrict__ src,
                                            int ld, int row0, int rows,
                                            int rowclamp,
                                            float (*dst)[LDT],
                                            int k0, int tid) {
    const int chunks = rows * (BK / 4);        // 16B chunks in the tile
    for (int idx = tid; idx < chunks; idx += 256) {
        int r = idx >> 3;                      // BK/4 == 8 chunks per row
        int q = (idx & 7) * 4;
        int gr = row0 + r;
        if (gr >= rowclamp) gr = rowclamp - 1; // clamp: garbage cols are dead
        __builtin_amdgcn_global_load_async_to_lds_b128(
            (v4i*)(src + (long long)gr * ld + k0 + q),
            (v4i*)&dst[r][q], 0, 0);
    }
}
#else
#define NBUF 1
#endif

__global__ __launch_bounds__(256)
void wmma_gemm_kernel(const float* __restrict__ Amat, int lda,
                      const float* __restrict__ Bmat, int ldb, int Brows,
                      int K, int mode,
                      const float* __restrict__ bias,     // mode 0
                      const float* __restrict__ ybase,    // mode 1
                      const float* __restrict__ t3,       // mode 1
                      const int*   __restrict__ ratio_p,  // mode 1
                      float* __restrict__ out, int Cdim) {
    __shared__ float lsA[NBUF][BM][LDT];
    __shared__ float lsB[NBUF][BN][LDT];

    const int tid  = threadIdx.x;
    const int lane = tid & 31;
    const int wave = tid >> 5;      // 0..7
    const int wm   = wave & 3;      // wave M index 0..3  (4 * 16 = 64)
    const int wn   = wave >> 2;     // wave N index 0..1  (2 * 64 = 128)
    const int h    = lane >> 4;     // lane half
    const int mr   = lane & 15;     // row/col within fragment

    const int row0 = blockIdx.y * BM;
    const int col0 = blockIdx.x * BN;

    v8f acc[4] = {};                // 16 x 64 strip per wave

#if USE_ASYNC_LDS
    // -------- double-buffered async DMA staging --------
    int buf = 0;
    stage_async(Amat, lda, row0, BM, 0x7fffffff, lsA[0], 0, tid);
    stage_async(Bmat, ldb, col0, BN, Brows,      lsB[0], 0, tid);
    wait_async_zero();
    __syncthreads();

    for (int k0 = 0; k0 < K; k0 += BK) {
        if (k0 + BK < K) {   // kick off next chunk while computing this one
            stage_async(Amat, lda, row0, BM, 0x7fffffff, lsA[buf ^ 1], k0 + BK, tid);
            stage_async(Bmat, ldb, col0, BN, Brows,      lsB[buf ^ 1], k0 + BK, tid);
        }
#pragma unroll
        for (int kk = 0; kk < BK; kk += 4) {
            const int kf = kk + 2 * h;                        // 8B-aligned
            v2f a = *(const v2f*)&lsA[buf][wm * 16 + mr][kf];
#pragma unroll
            for (int i = 0; i < 4; ++i) {
                v2f b = *(const v2f*)&lsB[buf][wn * 64 + i * 16 + mr][kf];
                acc[i] = __builtin_amdgcn_wmma_f32_16x16x4_f32(
                    false, a, false, b, (short)0, acc[i], false, false);
            }
        }
        wait_async_zero();     // next buffer fully landed in LDS
        __syncthreads();       // all waves done reading current buffer
        buf ^= 1;
    }
#else
    // -------- synchronous staging fallback --------
    const int kc = tid & (BK - 1);
    const int r0 = tid >> 5;
    for (int k0 = 0; k0 < K; k0 += BK) {
#pragma unroll
        for (int i = 0; i < BM / 8; ++i) {
            int r = r0 + i * 8;
            lsA[0][r][kc] = Amat[(long long)(row0 + r) * lda + k0 + kc];
        }
#pragma unroll
        for (int i = 0; i < BN / 8; ++i) {
            int r = r0 + i * 8;
            int gr = col0 + r;
            lsB[0][r][kc] = (gr < Brows)
                          ? Bmat[(long long)gr * ldb + k0 + kc] : 0.0f;
        }
        if (k0 + BK < K) {
            __builtin_prefetch(&Amat[(long long)(row0 + r0) * lda + k0 + BK + kc], 0, 0);
            if (col0 + r0 < Brows)
                __builtin_prefetch(&Bmat[(long long)(col0 + r0) * ldb + k0 + BK + kc], 0, 0);
        }
        __syncthreads();
#pragma unroll
        for (int kk = 0; kk < BK; kk += 4) {
            const int kf = kk + 2 * h;
            v2f a = *(const v2f*)&lsA[0][wm * 16 + mr][kf];
#pragma unroll
            for (int i = 0; i < 4; ++i) {
                v2f b = *(const v2f*)&lsB[0][wn * 64 + i * 16 + mr][kf];
                acc[i] = __builtin_amdgcn_wmma_f32_16x16x4_f32(
                    false, a, false, b, (short)0, acc[i], false, false);
            }
        }
        __syncthreads();
    }
#endif

    // ---- epilogue ----
    float ratio = 1.0f;
    if (mode == 1) ratio = (float)ratio_p[0];

#pragma unroll
    for (int i = 0; i < 4; ++i) {
        int col = col0 + wn * 64 + i * 16 + mr;
        if (col < Cdim) {
#pragma unroll
            for (int r = 0; r < 8; ++r) {
                int row = row0 + wm * 16 + r + 8 * h;
                long long o = (long long)row * Cdim + col;
                float v = acc[i][r];
                if (mode == 0) {
                    out[o] = v + bias[col];
                } else {
                    out[o] = ybase[o] + 0.5f * ratio * (v + t3[row]);
                }
            }
        }
    }
}

// --------------------------- ISDA prep kernels -----------------------------

// B2 = [W^2 | W]  (C x 2A)
__global__ void prep_b2_kernel(const float* __restrict__ W,
                               float* __restrict__ B2) {
    int idx = blockIdx.x * blockDim.x + threadIdx.x;
    int stride = gridDim.x * blockDim.x;
    for (; idx < C_CLS * A_F; idx += stride) {
        int c = idx / A_F;
        int a = idx - c * A_F;
        float w = W[idx];
        long long base = (long long)c * (2 * A_F);
        B2[base + a]        = w * w;
        B2[base + A_F + a]  = w;
    }
}

// A2 = [cv_n | -2*w_n*cv_n]  (N x 2A),  t3[n] = sum_a w_n^2 * cv_n
__global__ __launch_bounds__(256)
void prep_a2_t3_kernel(const int* __restrict__ labels,
                       const float* __restrict__ ncov,
                       const float* __restrict__ W,
                       float* __restrict__ A2,
                       float* __restrict__ t3) {
    __shared__ float red[256];
    int n = blockIdx.x;
    int c = labels[n];
    long long src = (long long)c * A_F;
    long long dst = (long long)n * (2 * A_F);
    float local = 0.0f;
    for (int a = threadIdx.x; a < A_F; a += 256) {
        float cv = ncov[src + a];
        float w  = W[src + a];
        A2[dst + a]       = cv;
        A2[dst + A_F + a] = -2.0f * w * cv;
        local += w * w * cv;
    }
    red[threadIdx.x] = local;
    __syncthreads();
    for (int s = 128; s > 0; s >>= 1) {
        if (threadIdx.x < s) red[threadIdx.x] += red[threadIdx.x + s];
        __syncthreads();
    }
    if (threadIdx.x == 0) t3[n] = red[0];
}

// ------------------------- softmax + NLL loss ------------------------------

__global__ __launch_bounds__(256)
void loss_kernel(const float* __restrict__ isda,
                 const int* __restrict__ labels,
                 float* __restrict__ loss_out) {
    __shared__ float red[256];
    int n = blockIdx.x;
    const float* row = isda + (long long)n * C_CLS;

    float mx = -3.402823466e38f;
    for (int c = threadIdx.x; c < C_CLS; c += 256) mx = fmaxf(mx, row[c]);
    red[threadIdx.x] = mx;
    __syncthreads();
    for (int s = 128; s > 0; s >>= 1) {
        if (threadIdx.x < s)
            red[threadIdx.x] = fmaxf(red[threadIdx.x], red[threadIdx.x + s]);
        __syncthreads();
    }
    mx = red[0];
    __syncthreads();

    float sum = 0.0f;
    for (int c = threadIdx.x; c < C_CLS; c += 256) sum += expf(row[c] - mx);
    red[threadIdx.x] = sum;
    __syncthreads();
    for (int s = 128; s > 0; s >>= 1) {
        if (threadIdx.x < s) red[threadIdx.x] += red[threadIdx.x + s];
        __syncthreads();
    }
    if (threadIdx.x == 0) {
        float lp = row[labels[n]] - mx - logf(red[0]);
        atomicAdd(loss_out, -lp / (float)N_S);
    }
}

// ------------------------------ launcher -----------------------------------

extern "C" void kernel_launch(void* const* d_in, const int* in_sizes, int n_in,
                              void* d_out, int out_size, void* d_ws, size_t ws_size,
                              hipStream_t stream) {
    (void)in_sizes; (void)n_in; (void)out_size; (void)ws_size;

    const float* features = (const float*)d_in[0];
    const int*   labels   = (const int*)  d_in[1];
    const float* fc_w     = (const float*)d_in[2];
    const float* fc_b     = (const float*)d_in[3];
    const float* ave_st   = (const float*)d_in[4];
    const float* cov_st   = (const float*)d_in[5];
    const float* amt_st   = (const float*)d_in[6];
    const int*   ratio_p  = (const int*)  d_in[7];

    // d_out layout: [loss(1), y(512*1000), new_ave(C*A), new_cov(C*A), new_amount(C)]
    float* out    = (float*)d_out;
    float* o_loss = out;
    float* o_y    = out + 1;
    float* o_ave  = o_y + (long long)N_S * C_CLS;
    float* o_cov  = o_ave + (long long)C_CLS * A_F;
    float* o_amt  = o_cov + (long long)C_CLS * A_F;

    // workspace layout (floats)
    float* ws       = (float*)d_ws;
    float* w_counts = ws;                                   // 1024
    float* w_t3     = ws + 1024;                            // 512
    float* w_A2     = ws + 1536;                            // N * 2A
    float* w_B2     = w_A2 + (long long)N_S * 2 * A_F;      // C * 2A
    float* w_isda   = w_B2 + (long long)C_CLS * 2 * A_F;    // N * C

    // 1) zero accumulators (new_ave/new_cov regions double as sum / sq-sum)
    fillzero_kernel<<<2048, 256, 0, stream>>>(o_ave, C_CLS * A_F);
    fillzero_kernel<<<2048, 256, 0, stream>>>(o_cov, C_CLS * A_F);
    fillzero_kernel<<<8, 256, 0, stream>>>(w_counts, 1536);
    fillzero_kernel<<<1, 64, 0, stream>>>(o_loss, 1);

    // 2) segment statistics
    counts_kernel<<<2, 256, 0, stream>>>(labels, w_counts);
    sums_kernel<<<(N_S * A_F) / 256, 256, 0, stream>>>(features, labels, o_ave);
    sq_kernel<<<2048, 256, 0, stream>>>(features, labels, w_counts, o_ave, o_cov);
    finalize_cv_kernel<<<2048, 256, 0, stream>>>(w_counts, ave_st, cov_st, amt_st,
                                                 o_ave, o_cov, o_amt);

    // 3) GEMM 1: y = features @ W^T + b   (M=512, N=1000, K=2048)
    dim3 grid((C_CLS + BN - 1) / BN, N_S / BM);
    wmma_gemm_kernel<<<grid, 256, 0, stream>>>(
        features, A_F, fc_w, A_F, C_CLS, A_F, /*mode=*/0,
        fc_b, nullptr, nullptr, nullptr, o_y, C_CLS);

    // 4) ISDA prep: A2=[cv | -2 w cv], B2=[W^2 | W], t3
    prep_b2_kernel<<<2048, 256, 0, stream>>>(fc_w, w_B2);
    prep_a2_t3_kernel<<<N_S, 256, 0, stream>>>(labels, o_cov, fc_w, w_A2, w_t3);

    // 5) GEMM 2 (K=4096): isda = y + 0.5*ratio*(A2@B2^T + t3)
    wmma_gemm_kernel<<<grid, 256, 0, stream>>>(
        w_A2, 2 * A_F, w_B2, 2 * A_F, C_CLS, 2 * A_F, /*mode=*/1,
        nullptr, o_y, w_t3, ratio_p, w_isda, C_CLS);

    // 6) log-softmax + NLL mean
    loss_kernel<<<N_S, 256, 0, stream>>>(w_isda, labels, o_loss);
}